// LATGCN_52475910423196
// MI455X (gfx1250) — compile-verified
//
#include <hip/hip_runtime.h>

#define N_NODES 40000
#define N_EDGES 640000
#define F_IN    64
#define F_OUT   128
#define PERIODS 12
#define GDIM    512   // 4 gates * F_OUT
#define KDIM    192   // F_IN + F_OUT
#define RPB     64    // rows per block in step kernel

typedef __attribute__((ext_vector_type(2))) float v2f;
typedef __attribute__((ext_vector_type(8))) float v8f;
typedef __attribute__((__vector_size__(4 * sizeof(int)))) int v4i;

#define AS1 __attribute__((address_space(1)))
#define AS3 __attribute__((address_space(3)))

#if __has_builtin(__builtin_amdgcn_global_load_async_to_lds_b128)
#define HAVE_ASYNC_LDS 1
#else
#define HAVE_ASYNC_LDS 0
#endif

#if HAVE_ASYNC_LDS
__device__ __forceinline__ void async_copy16(const float* gsrc, float* ldst) {
    __builtin_amdgcn_global_load_async_to_lds_b128(
        (AS1 v4i*)gsrc, (AS3 v4i*)ldst, 0, 0);
}
#endif

__device__ __forceinline__ void wait_async_zero() {
#if __has_builtin(__builtin_amdgcn_s_wait_asynccnt)
    __builtin_amdgcn_s_wait_asynccnt(0);
#else
    asm volatile("s_wait_asynccnt 0" ::: "memory");
#endif
}

struct GateW {
    const float* Wx[4];
    const float* bx[4];
    const float* Wh[4];
    const float* bh[4];
};

// ---------------------------------------------------------------- utilities
__global__ void fill_f32(float* __restrict__ p, float v, int n) {
    int i = blockIdx.x * blockDim.x + threadIdx.x;
    if (i < n) p[i] = v;
}

__global__ void accum_deg(const int* __restrict__ cols,
                          const float* __restrict__ w,
                          float* __restrict__ deg, int e_cnt) {
    int e = blockIdx.x * blockDim.x + threadIdx.x;
    if (e < e_cnt) atomicAdd(&deg[cols[e]], w[e]);
}

__global__ void finalize_dinv(float* __restrict__ deg, int n) {
    int i = blockIdx.x * blockDim.x + threadIdx.x;
    if (i < n) {
        float d = deg[i];
        deg[i] = (d > 0.0f) ? rsqrtf(d) : 0.0f;   // deg becomes dinv in place
    }
}

__global__ void edge_norm(const int* __restrict__ rows,
                          const int* __restrict__ cols,
                          const float* __restrict__ w,
                          const float* __restrict__ dinv,
                          float* __restrict__ norm, int e_cnt) {
    int e = blockIdx.x * blockDim.x + threadIdx.x;
    if (e < e_cnt) norm[e] = dinv[rows[e]] * w[e] * dinv[cols[e]];
}

// X[n][f][t] (t fastest) -> XT[t][n][f]
__global__ void transpose_X(const float* __restrict__ X, float* __restrict__ XT) {
    int i = blockIdx.x * blockDim.x + threadIdx.x;   // over N*F_IN
    if (i >= N_NODES * F_IN) return;
    const float* src = X + (size_t)i * PERIODS;
#pragma unroll
    for (int t = 0; t < PERIODS; ++t)
        XT[(size_t)t * N_NODES * F_IN + i] = src[t];
}

// Build Ball[192][512]: rows 0..63  = W_x[g] @ W_h[g][0:128]   (folded GCN weight)
//                       rows 64..191= W_h[g][128:256]          (H path)
// bias[512] = b_x[g] @ W_h[g][0:128] + b_h[g]
__global__ void prep_weights(GateW gw, float* __restrict__ Ball,
                             float* __restrict__ bias) {
    int idx = blockIdx.x * blockDim.x + threadIdx.x;
    if (idx >= GDIM * (KDIM + 1)) return;
    int j = idx % GDIM;
    int r = idx / GDIM;
    int g = j >> 7, jj = j & 127;
    if (r < F_IN) {
        const float* Wx = gw.Wx[g];
        const float* Wh = gw.Wh[g];
        float s = 0.0f;
        for (int k = 0; k < F_OUT; ++k)
            s += Wx[r * F_OUT + k] * Wh[k * F_OUT + jj];
        Ball[(size_t)r * GDIM + j] = s;
    } else if (r < KDIM) {
        int k = r - F_IN;
        Ball[(size_t)r * GDIM + j] = gw.Wh[g][(size_t)(F_OUT + k) * F_OUT + jj];
    } else {
        const float* Wh = gw.Wh[g];
        float s = gw.bh[g][jj];
        for (int k = 0; k < F_OUT; ++k)
            s += gw.bx[g][k] * Wh[k * F_OUT + jj];
        bias[j] = s;
    }
}

__global__ void softmax_att(const float* __restrict__ att, float* __restrict__ probs) {
    if (blockIdx.x == 0 && threadIdx.x == 0) {
        float mx = -1e30f;
        for (int t = 0; t < PERIODS; ++t) mx = fmaxf(mx, att[t]);
        float e[PERIODS], s = 0.0f;
        for (int t = 0; t < PERIODS; ++t) { e[t] = expf(att[t] - mx); s += e[t]; }
        float inv = 1.0f / s;
        for (int t = 0; t < PERIODS; ++t) probs[t] = e[t] * inv;
    }
}

// AXt[n][f] = dinv[n]^2 * XT_t[n][f]   (self-loop term of A_norm)
__global__ void ax_init(const float* __restrict__ XTt,
                        const float* __restrict__ dinv,
                        float* __restrict__ AXt) {
    int i = blockIdx.x * blockDim.x + threadIdx.x;
    if (i >= N_NODES * F_IN) return;
    float d = dinv[i >> 6];
    AXt[i] = d * d * XTt[i];
}

// edge scatter: AXt[col] += norm[e] * XT_t[row]   (64 lanes per edge)
__global__ void ax_scatter(const int* __restrict__ rows,
                           const int* __restrict__ cols,
                           const float* __restrict__ norm,
                           const float* __restrict__ XTt,
                           float* __restrict__ AXt) {
    unsigned i = blockIdx.x * blockDim.x + threadIdx.x;   // E*64 threads
    if (i >= (unsigned)N_EDGES * F_IN) return;
    int e = (int)(i >> 6);
    int f = (int)(i & 63);
    float v = norm[e] * XTt[(size_t)rows[e] * F_IN + f];
    atomicAdd(&AXt[(size_t)cols[e] * F_IN + f], v);
}

// ------------------------------------------------------- fused GEMM + gates
// One block = 64 output rows, 8 waves. Wave w owns jj-tile w*16 across all
// four gates (cols jj, jj+128, jj+256, jj+384) and all 4 row-tiles, so each
// B fragment feeds 4 WMMAs and the LSTM nonlinearity runs from registers.
__global__ void __launch_bounds__(256, 1)
step_kernel(const float* __restrict__ AXt,
            const float* __restrict__ Hread,
            float* __restrict__ Hwrite,
            float* __restrict__ Cst,
            const float* __restrict__ Ball,
            const float* __restrict__ bias,
            const float* __restrict__ probs,
            int t,
            float* __restrict__ Hacc) {
    __shared__ float Alds[RPB][KDIM];   // 48 KB
    const int tid  = threadIdx.x;
    const int row0 = blockIdx.x * RPB;

    // ---- stage A = [AXt rows | H rows], 64 x 192 ----
#if HAVE_ASYNC_LDS
    // AXt half: 64 rows x 64 floats = 1024 16B chunks (16 per row)
    for (int ch = tid; ch < RPB * (F_IN / 4); ch += 256) {
        int m  = ch >> 4;
        int k4 = (ch & 15) << 2;
        async_copy16(AXt + (size_t)(row0 + m) * F_IN + k4, &Alds[m][k4]);
    }
    // H half: 64 rows x 128 floats = 2048 16B chunks (32 per row)
    for (int ch = tid; ch < RPB * (F_OUT / 4); ch += 256) {
        int m  = ch >> 5;
        int k4 = (ch & 31) << 2;
        async_copy16(Hread + (size_t)(row0 + m) * F_OUT + k4, &Alds[m][F_IN + k4]);
    }
    wait_async_zero();
#else
    for (int i = tid; i < RPB * KDIM; i += 256) {
        int m = i / KDIM, k = i % KDIM;
        float v = (k < F_IN)
                      ? AXt[(size_t)(row0 + m) * F_IN + k]
                      : Hread[(size_t)(row0 + m) * F_OUT + (k - F_IN)];
        Alds[m][k] = v;
    }
#endif
    __syncthreads();

    const int lane  = tid & 31;
    const int wave  = tid >> 5;
    const int l15   = lane & 15;
    const int khalf = (lane >> 4) << 1;     // 0 for lanes 0-15, 2 for 16-31
    const int mrow  = (lane >> 4) << 3;     // C/D row offset per lane half
    const int jjb   = wave * 16;            // this wave's jj tile

    v8f acc[4][4] = {};                     // [row_tile][gate]

    for (int kb = 0; kb < KDIM; kb += 4) {
        v2f a[4], b[4];
#pragma unroll
        for (int rt = 0; rt < 4; ++rt) {
            a[rt].x = Alds[rt * 16 + l15][kb + khalf];
            a[rt].y = Alds[rt * 16 + l15][kb + khalf + 1];
        }
#pragma unroll
        for (int g = 0; g < 4; ++g) {
            const float* bp = Ball + (size_t)(kb + khalf) * GDIM + g * F_OUT + jjb + l15;
            b[g].x = bp[0];
            b[g].y = bp[GDIM];
        }
#pragma unroll
        for (int rt = 0; rt < 4; ++rt)
#pragma unroll
            for (int g = 0; g < 4; ++g)
                acc[rt][g] = __builtin_amdgcn_wmma_f32_16x16x4_f32(
                    false, a[rt], false, b[g], (short)0, acc[rt][g], false, false);
    }

    float bv[4];
#pragma unroll
    for (int g = 0; g < 4; ++g) bv[g] = bias[g * F_OUT + jjb + l15];
    const float p = probs[t];

#pragma unroll
    for (int rt = 0; rt < 4; ++rt) {
#pragma unroll
        for (int r = 0; r < 8; ++r) {
            int row     = row0 + rt * 16 + mrow + r;
            size_t gidx = (size_t)row * F_OUT + jjb + l15;
            float I  = 1.0f / (1.0f + expf(-(acc[rt][0][r] + bv[0])));
            float Fg = 1.0f / (1.0f + expf(-(acc[rt][1][r] + bv[1])));
            float Ct = tanhf(acc[rt][2][r] + bv[2]);
            float O  = 1.0f / (1.0f + expf(-(acc[rt][3][r] + bv[3])));
            float cn = Fg * Cst[gidx] + I * Ct;
            Cst[gidx] = cn;
            float h = O * tanhf(cn);
            Hwrite[gidx] = h;
            Hacc[gidx] += p * h;
        }
    }
}

// ---------------------------------------------------------------- launcher
static inline int gridFor(long n, int b) { return (int)((n + b - 1) / b); }

extern "C" void kernel_launch(void* const* d_in, const int* in_sizes, int n_in,
                              void* d_out, int out_size, void* d_ws, size_t ws_size,
                              hipStream_t stream) {
    const float* X    = (const float*)d_in[0];
    const int*   eidx = (const int*)d_in[1];
    const int*   erow = eidx;             // edge_index[0] = sources
    const int*   ecol = eidx + N_EDGES;   // edge_index[1] = targets
    const float* ew   = (const float*)d_in[2];
    const float* att  = (const float*)d_in[3];

    GateW gw;
    for (int g = 0; g < 4; ++g) {
        gw.Wx[g] = (const float*)d_in[4 + 4 * g];
        gw.bx[g] = (const float*)d_in[5 + 4 * g];
        gw.Wh[g] = (const float*)d_in[6 + 4 * g];
        gw.bh[g] = (const float*)d_in[7 + 4 * g];
    }

    // workspace layout (float offsets)
    float* ws    = (float*)d_ws;
    float* probs = ws + 0;                 // 16
    float* bias  = ws + 64;                // 512
    float* Ball  = ws + 576;               // 192*512 = 98304
    float* dinv  = ws + 98880;             // N (deg, then dinv in place)
    float* norm  = ws + 138880;            // E
    float* XT    = ws + 778880;            // 12*N*64
    float* AXt   = ws + 31498880;          // N*64
    float* Hb0   = ws + 34058880;          // N*128
    float* Hb1   = ws + 39178880;          // N*128
    float* Cst   = ws + 44298880;          // N*128  (total ~198 MB)
    float* Hacc  = (float*)d_out;          // N*128 output accumulator

    // init
    fill_f32<<<gridFor(N_NODES, 256), 256, 0, stream>>>(dinv, 1.0f, N_NODES); // self-loop deg
    fill_f32<<<gridFor((long)N_NODES * F_OUT, 256), 256, 0, stream>>>(Hb0, 0.0f, N_NODES * F_OUT);
    fill_f32<<<gridFor((long)N_NODES * F_OUT, 256), 256, 0, stream>>>(Cst, 0.0f, N_NODES * F_OUT);
    fill_f32<<<gridFor((long)N_NODES * F_OUT, 256), 256, 0, stream>>>(Hacc, 0.0f, N_NODES * F_OUT);

    // graph normalization
    accum_deg<<<gridFor(N_EDGES, 256), 256, 0, stream>>>(ecol, ew, dinv, N_EDGES);
    finalize_dinv<<<gridFor(N_NODES, 256), 256, 0, stream>>>(dinv, N_NODES);
    edge_norm<<<gridFor(N_EDGES, 256), 256, 0, stream>>>(erow, ecol, ew, dinv, norm, N_EDGES);

    // preprocessing
    transpose_X<<<gridFor((long)N_NODES * F_IN, 256), 256, 0, stream>>>(X, XT);
    prep_weights<<<gridFor((long)GDIM * (KDIM + 1), 256), 256, 0, stream>>>(gw, Ball, bias);
    softmax_att<<<1, 32, 0, stream>>>(att, probs);

    // recurrent time loop
    for (int t = 0; t < PERIODS; ++t) {
        const float* XTt = XT + (size_t)t * N_NODES * F_IN;
        ax_init<<<gridFor((long)N_NODES * F_IN, 256), 256, 0, stream>>>(XTt, dinv, AXt);
        ax_scatter<<<gridFor((long)N_EDGES * F_IN, 256), 256, 0, stream>>>(
            erow, ecol, norm, XTt, AXt);
        const float* Hr = (t & 1) ? Hb1 : Hb0;
        float*       Hw = (t & 1) ? Hb0 : Hb1;
        step_kernel<<<N_NODES / RPB, 256, 0, stream>>>(
            AXt, Hr, Hw, Cst, Ball, bias, probs, t, Hacc);
    }
}